// MaxSimPartition_30812095381662
// MI455X (gfx1250) — compile-verified
//
#include <hip/hip_runtime.h>

// ---------------------------------------------------------------------------
// MaxSim partition scoring for MI455X (gfx1250), f32 path via
// V_WMMA_F32_16X16X4_F32 to match the f32 reference numerics exactly.
// ---------------------------------------------------------------------------

typedef __attribute__((ext_vector_type(2))) float v2f;
typedef __attribute__((ext_vector_type(8))) float v8f;

#define N_DOCS  10000
#define DOC_LEN 128
#define DIM     128
#define BATCH   16
#define QLEN    32
#define KPIDS   1024
#define TOPK    100

#define WAVES_PER_WG 8
#define WG_PER_ROW   32

// ---------------- kernel 1: per-row unique(pids), -1 padded ----------------
__global__ __launch_bounds__(KPIDS)
void uniq_kernel(const int* __restrict__ pids, const int* __restrict__ boundaries,
                 int* __restrict__ upids) {
  __shared__ int s[KPIDS];
  const int b = blockIdx.x, tid = threadIdx.x;
  int p = pids[b * KPIDS + tid] - boundaries[0];
  if (p < 0 || p >= N_DOCS) p = -1;
  s[tid] = p;
  // descending bitonic sort
  for (int k = 2; k <= KPIDS; k <<= 1)
    for (int j = k >> 1; j > 0; j >>= 1) {
      __syncthreads();
      int ixj = tid ^ j;
      if (ixj > tid) {
        int a = s[tid], c = s[ixj];
        bool desc_block = ((tid & k) == 0);
        if (desc_block ? (a < c) : (a > c)) { s[tid] = c; s[ixj] = a; }
      }
    }
  __syncthreads();
  int v = s[tid];
  int prev = (tid == 0) ? 0x7fffffff : s[tid - 1];
  __syncthreads();
  if (tid > 0 && v == prev) s[tid] = -1;   // mark duplicates
  // second descending sort pushes the -1s to the end
  for (int k = 2; k <= KPIDS; k <<= 1)
    for (int j = k >> 1; j > 0; j >>= 1) {
      __syncthreads();
      int ixj = tid ^ j;
      if (ixj > tid) {
        int a = s[tid], c = s[ixj];
        bool desc_block = ((tid & k) == 0);
        if (desc_block ? (a < c) : (a > c)) { s[tid] = c; s[ixj] = a; }
      }
    }
  __syncthreads();
  upids[b * KPIDS + tid] = s[tid];
}

// ---------------- kernel 2: WMMA MaxSim scores, one wave per doc ------------
__global__ __launch_bounds__(WAVES_PER_WG * 32)
void maxsim_kernel(const float* __restrict__ q_vectors,
                   const float* __restrict__ vectors,
                   const int*   __restrict__ upids,
                   float*       __restrict__ scores) {
  __shared__ float qs[QLEN * DIM];           // 16 KB, one batch row of queries
  const int b   = blockIdx.x;
  const int tid = threadIdx.x;

  // stage Q[b] (32x128 f32) into LDS, vectorized
  const float4* qsrc = (const float4*)(q_vectors + (size_t)b * QLEN * DIM);
  float4* qdst = (float4*)qs;
  for (int i = tid; i < QLEN * DIM / 4; i += WAVES_PER_WG * 32) qdst[i] = qsrc[i];
  __syncthreads();

  const int lane = tid & 31;
  const int wave = tid >> 5;
  const int nlo  = lane & 15;           // M (A) or N (B) within the tile
  const int koff = (lane >> 4) * 2;     // K pair offset per ISA 16x4 f32 layout
  const int gw   = blockIdx.y * WAVES_PER_WG + wave;
  const int nwaves = WG_PER_ROW * WAVES_PER_WG;

  const float* qm0 = qs + nlo * DIM + koff;          // A tile: queries 0..15
  const float* qm1 = qs + (nlo + 16) * DIM + koff;   // A tile: queries 16..31

  for (int i = gw; i < KPIDS; i += nwaves) {
    const int pid = upids[b * KPIDS + i];            // wave-uniform
    float score;
    if (pid < 0) {
      score = -__builtin_inff();
    } else {
      const float* doc  = vectors + (size_t)pid * (DOC_LEN * DIM);
      const float* dcol = doc + nlo * DIM + koff;    // B = doc^T, lane streams its token
      float m0[8], m1[8];
      #pragma unroll
      for (int r = 0; r < 8; ++r) { m0[r] = -__builtin_inff(); m1[r] = -__builtin_inff(); }

      for (int nt = 0; nt < DOC_LEN / 16; ++nt) {    // 8 tiles of doc tokens
        v8f c0 = {}; v8f c1 = {};
        const float* dp = dcol + nt * 16 * DIM;
        #pragma unroll
        for (int kk = 0; kk < DIM / 4; ++kk) {       // K=128 in 32 chained WMMAs
          v2f a0 = *(const v2f*)(qm0 + kk * 4);
          v2f a1 = *(const v2f*)(qm1 + kk * 4);
          v2f bb = *(const v2f*)(dp  + kk * 4);
          c0 = __builtin_amdgcn_wmma_f32_16x16x4_f32(false, a0, false, bb,
                                                     (short)0, c0, false, false);
          c1 = __builtin_amdgcn_wmma_f32_16x16x4_f32(false, a1, false, bb,
                                                     (short)0, c1, false, false);
        }
        #pragma unroll
        for (int r = 0; r < 8; ++r) {
          m0[r] = fmaxf(m0[r], c0[r]);
          m1[r] = fmaxf(m1[r], c1[r]);
        }
      }
      // max over doc tokens: reduce across the 16 lanes of each half (wave32)
      #pragma unroll
      for (int mask = 1; mask <= 8; mask <<= 1) {
        #pragma unroll
        for (int r = 0; r < 8; ++r) {
          m0[r] = fmaxf(m0[r], __shfl_xor(m0[r], mask, 32));
          m1[r] = fmaxf(m1[r], __shfl_xor(m1[r], mask, 32));
        }
      }
      // mean over the 32 queries: 8 rows per lane half, fold halves with xor 16
      float ssum = 0.f;
      #pragma unroll
      for (int r = 0; r < 8; ++r) ssum += m0[r] + m1[r];
      ssum += __shfl_xor(ssum, 16, 32);
      score = ssum * (1.0f / QLEN);
    }
    if (lane == 0) scores[b * KPIDS + i] = score;
  }
}

// ---------------- kernel 3: top-100 per row (bitonic, tie -> lower idx) -----
__global__ __launch_bounds__(KPIDS)
void topk_kernel(const float* __restrict__ scores, const int* __restrict__ upids,
                 float* __restrict__ out_scores, int* __restrict__ out_pids) {
  __shared__ float key[KPIDS];
  __shared__ int   idx[KPIDS];
  const int b = blockIdx.x, tid = threadIdx.x;
  key[tid] = scores[b * KPIDS + tid];
  idx[tid] = tid;
  for (int k = 2; k <= KPIDS; k <<= 1)
    for (int j = k >> 1; j > 0; j >>= 1) {
      __syncthreads();
      int ixj = tid ^ j;
      if (ixj > tid) {
        float ka = key[tid], kc = key[ixj];
        int   ia = idx[tid], ic = idx[ixj];
        bool a_first = (ka > kc) || (ka == kc && ia < ic); // desired desc order
        bool desc_block = ((tid & k) == 0);
        bool do_swap = desc_block ? !a_first : a_first;
        if (do_swap) { key[tid] = kc; key[ixj] = ka; idx[tid] = ic; idx[ixj] = ia; }
      }
    }
  __syncthreads();
  if (tid < TOPK) {
    out_scores[b * TOPK + tid] = key[tid];
    out_pids  [b * TOPK + tid] = upids[b * KPIDS + idx[tid]];
  }
}

// ---------------------------------------------------------------------------
extern "C" void kernel_launch(void* const* d_in, const int* in_sizes, int n_in,
                              void* d_out, int out_size, void* d_ws, size_t ws_size,
                              hipStream_t stream) {
  const float* q    = (const float*)d_in[0];   // [16,32,128] f32
  const float* vec  = (const float*)d_in[1];   // [10000,128,128] f32
  const int*   pids = (const int*)d_in[2];     // [16,1024] i32
  const int*   bnd  = (const int*)d_in[3];     // [2] i32
  (void)in_sizes; (void)n_in; (void)out_size; (void)ws_size;

  int*   upids  = (int*)d_ws;                                        // 64 KB
  float* scores = (float*)((char*)d_ws + BATCH * KPIDS * sizeof(int)); // 64 KB

  float* out_scores = (float*)d_out;                 // [16,100] f32
  int*   out_pids   = (int*)d_out + BATCH * TOPK;    // [16,100] i32 (bit pattern)

  uniq_kernel<<<BATCH, KPIDS, 0, stream>>>(pids, bnd, upids);
  maxsim_kernel<<<dim3(BATCH, WG_PER_ROW), WAVES_PER_WG * 32, 0, stream>>>(
      q, vec, upids, scores);
  topk_kernel<<<BATCH, KPIDS, 0, stream>>>(scores, upids, out_scores, out_pids);
}